// NeuroGenesis_55267639164946
// MI455X (gfx1250) — compile-verified
//
#include <hip/hip_runtime.h>

#define MN 8192
#define ME 16384
#define FD 256
#define FDE 128

typedef __attribute__((ext_vector_type(2))) float v2f;
typedef __attribute__((ext_vector_type(8))) float v8f;

// ---------------------------------------------------------------------------
// K1: probs / divs via V_WMMA_F32_16X16X4_F32.
// One wave32 handles 16 node rows. A = 16x4 tile of `nodes` (documented f32
// A layout: lanes 0-15 hold K=0,1; lanes 16-31 hold K=2,3). B = 4x16 with
// W replicated across all 16 columns (half-split K layout mirroring A/C/D).
// D[m][n] == dot(nodes[m0+m], W[k-chunk]) for every n; accumulate over K=256.
// ---------------------------------------------------------------------------
__global__ void ng_probs_kernel(const float* __restrict__ nodes,
                                const float* __restrict__ W,
                                const float* __restrict__ bias_p,
                                const float* __restrict__ u_div,
                                const float* __restrict__ active_nodes,
                                int* __restrict__ divs) {
    const int lane = threadIdx.x & 31;
    const int wave = blockIdx.x * (blockDim.x >> 5) + (threadIdx.x >> 5);
    const int m0 = wave * 16;
    if (m0 >= MN) return;                         // uniform per wave

    const int half = lane >> 4;                   // 0: lanes 0-15, 1: lanes 16-31
    const int lm   = lane & 15;                   // A-matrix row within the tile
    const float* __restrict__ arow = nodes + (size_t)(m0 + lm) * FD + half * 2;
    const float* __restrict__ wofs = W + half * 2;

    v8f c = {0.f, 0.f, 0.f, 0.f, 0.f, 0.f, 0.f, 0.f};
    for (int k0 = 0; k0 < FD; k0 += 4) {
        v2f a, bm;
        a.x  = arow[k0 + 0];                      // A: K = k0 + 2*half + {0,1}
        a.y  = arow[k0 + 1];
        bm.x = wofs[k0 + 0];                      // B: same K, replicated over N
        bm.y = wofs[k0 + 1];
        c = __builtin_amdgcn_wmma_f32_16x16x4_f32(false, a, false, bm,
                                                  (short)0, c, false, false);
    }

    // D layout: VGPR r, lanes 0-15 -> M=r (N=lane); lanes 16-31 -> M=8+r.
    // Every column is identical, so lane L<8 uses c[L] for node m0+L and
    // lane 16+j uses c[j] for node m0+8+j: 16 lanes emit the 16 results.
    int m = -1;
    float dot = 0.0f;
    if (lm < 8) {
        m = m0 + half * 8 + lm;
        dot = c[lm];
    }
    if (m >= 0) {
        const float prob = 1.0f / (1.0f + expf(-(dot + bias_p[0])));
        const int dv = (active_nodes[m] != 0.0f) && (u_div[m] < prob);
        divs[m] = dv;
    }
}

// ---------------------------------------------------------------------------
// K2: single-block: count active nodes/edges, exclusive scan of divs,
// clip n_divs.  counters = {n_active, n_edges, n_divs, raw}.
// ---------------------------------------------------------------------------
__global__ void ng_scan_kernel(const float* __restrict__ active_nodes,
                               const float* __restrict__ active_edges,
                               const int* __restrict__ divs,
                               int* __restrict__ ranks,
                               int* __restrict__ counters) {
    __shared__ int sh[1024];
    __shared__ int carry, s_nact, s_nedge;
    const int t = threadIdx.x;

    int acc = 0;
    for (int i = t; i < MN; i += 1024) acc += (active_nodes[i] != 0.0f);
    sh[t] = acc;
    __syncthreads();
    for (int s = 512; s > 0; s >>= 1) {
        if (t < s) sh[t] += sh[t + s];
        __syncthreads();
    }
    if (t == 0) s_nact = sh[0];
    __syncthreads();

    acc = 0;
    for (int i = t; i < ME; i += 1024) acc += (active_edges[i] != 0.0f);
    sh[t] = acc;
    __syncthreads();
    for (int s = 512; s > 0; s >>= 1) {
        if (t < s) sh[t] += sh[t + s];
        __syncthreads();
    }
    if (t == 0) { s_nedge = sh[0]; carry = 0; }
    __syncthreads();

    // exclusive scan of divs[MN] in 1024-wide tiles (Hillis–Steele)
    for (int base = 0; base < MN; base += 1024) {
        const int v = divs[base + t];
        sh[t] = v;
        __syncthreads();
        for (int off = 1; off < 1024; off <<= 1) {
            const int add = (t >= off) ? sh[t - off] : 0;
            __syncthreads();
            sh[t] += add;
            __syncthreads();
        }
        ranks[base + t] = carry + sh[t] - v;      // exclusive rank
        __syncthreads();
        if (t == 1023) carry += sh[1023];
        __syncthreads();
    }

    if (t == 0) {
        const int raw = carry;
        const int allowed = MN - s_nact - 1;
        int nd = raw < 0 ? 0 : raw;
        if (nd > allowed) nd = allowed;
        counters[0] = s_nact;
        counters[1] = s_nedge;
        counters[2] = nd;
        counters[3] = raw;
    }
}

// ---------------------------------------------------------------------------
// K3: scatter node id of r-th dividing node into send_ids[n_edges + r]
// (segment_sum semantics: OOB dropped, add into zeroed buffer).
// ---------------------------------------------------------------------------
__global__ void ng_scatter_kernel(const int* __restrict__ divs,
                                  const int* __restrict__ ranks,
                                  const int* __restrict__ counters,
                                  int* __restrict__ send_ids) {
    const int i = blockIdx.x * blockDim.x + threadIdx.x;
    if (i >= MN) return;
    if (divs[i]) {
        const int tgt = counters[1] + ranks[i];
        if (tgt >= 0 && tgt < ME) atomicAdd(&send_ids[tgt], i);
    }
}

// ---------------------------------------------------------------------------
// K4: new_nodes = nodes + noise * mask_new  (B128 streaming), plus nanodes.
// ---------------------------------------------------------------------------
__global__ void ng_nodes_out_kernel(const float* __restrict__ nodes,
                                    const float* __restrict__ noise,
                                    const int* __restrict__ counters,
                                    float* __restrict__ out_nodes,
                                    float* __restrict__ out_nanodes) {
    const int nact = counters[0];
    const int ndiv = counters[2];
    const int idx = blockIdx.x * blockDim.x + threadIdx.x;   // MN * 64 float4s
    if (idx >= MN * (FD / 4)) return;
    const int row = idx >> 6;
    const bool newrow = (row >= nact) && (row < nact + ndiv);

    float4 v = ((const float4*)nodes)[idx];
    if (newrow) {
        const float4 n = ((const float4*)noise)[idx];
        v.x += n.x; v.y += n.y; v.z += n.z; v.w += n.w;
    }
    ((float4*)out_nodes)[idx] = v;

    if ((idx & 63) == 0) {
        int L = nact + ndiv; if (L > MN) L = MN;
        out_nanodes[row] = (row < L && row != MN - 1) ? 1.0f : 0.0f;
    }
}

// ---------------------------------------------------------------------------
// K5: new_edges = edges + noise * mask_new, plus nrec / nsend / naedges.
// ---------------------------------------------------------------------------
__global__ void ng_edges_out_kernel(const float* __restrict__ edges,
                                    const float* __restrict__ noise,
                                    const int* __restrict__ receivers,
                                    const int* __restrict__ senders,
                                    const int* __restrict__ counters,
                                    const int* __restrict__ send_ids,
                                    float* __restrict__ out_edges,
                                    int* __restrict__ out_nrec,
                                    int* __restrict__ out_nsend,
                                    float* __restrict__ out_naedges) {
    const int nact  = counters[0];
    const int nedge = counters[1];
    const int ndiv  = counters[2];
    const int idx = blockIdx.x * blockDim.x + threadIdx.x;   // ME * 32 float4s
    if (idx >= ME * (FDE / 4)) return;
    const int row = idx >> 5;
    const bool newrow = (row >= nedge) && (row < nedge + ndiv);

    float4 v = ((const float4*)edges)[idx];
    if (newrow) {
        const float4 n = ((const float4*)noise)[idx];
        v.x += n.x; v.y += n.y; v.z += n.z; v.w += n.w;
    }
    ((float4*)out_edges)[idx] = v;

    if ((idx & 31) == 0) {
        out_nrec[row]  = newrow ? (nact + (row - nedge)) : receivers[row];
        out_nsend[row] = (newrow ? 0 : senders[row]) + send_ids[row];
        int L = nedge + ndiv; if (L > ME) L = ME;
        out_naedges[row] = (row < L && row != ME - 1) ? 1.0f : 0.0f;
    }
}

extern "C" void kernel_launch(void* const* d_in, const int* in_sizes, int n_in,
                              void* d_out, int out_size, void* d_ws, size_t ws_size,
                              hipStream_t stream) {
    const float* nodes        = (const float*)d_in[0];
    const float* edges        = (const float*)d_in[1];
    const int*   receivers    = (const int*)  d_in[2];
    const int*   senders      = (const int*)  d_in[3];
    const float* active_nodes = (const float*)d_in[4];
    const float* active_edges = (const float*)d_in[5];
    const float* W            = (const float*)d_in[6];
    const float* b            = (const float*)d_in[7];
    const float* u_div        = (const float*)d_in[8];
    const float* noise_nodes  = (const float*)d_in[9];
    const float* noise_edges  = (const float*)d_in[10];

    // workspace layout (re-initialized every call)
    int* divs     = (int*)d_ws;          // MN
    int* ranks    = divs + MN;           // MN
    int* send_ids = ranks + MN;          // ME
    int* counters = send_ids + ME;       // 8

    // output layout (flat, return order)
    float* out_nodes   = (float*)d_out;
    float* out_edges   = out_nodes + (size_t)MN * FD;
    int*   out_nrec    = (int*)(out_edges + (size_t)ME * FDE);
    int*   out_nsend   = out_nrec + ME;
    float* out_nanodes = (float*)(out_nsend + ME);
    float* out_naedges = out_nanodes + MN;

    hipMemsetAsync(send_ids, 0, ME * sizeof(int), stream);

    // MN/16 = 512 wave-tiles, 4 waves per 128-thread block -> 128 blocks
    ng_probs_kernel<<<MN / 16 / 4, 128, 0, stream>>>(nodes, W, b, u_div,
                                                     active_nodes, divs);
    ng_scan_kernel<<<1, 1024, 0, stream>>>(active_nodes, active_edges,
                                           divs, ranks, counters);
    ng_scatter_kernel<<<MN / 256, 256, 0, stream>>>(divs, ranks, counters,
                                                    send_ids);
    ng_nodes_out_kernel<<<(MN * (FD / 4)) / 256, 256, 0, stream>>>(
        nodes, noise_nodes, counters, out_nodes, out_nanodes);
    ng_edges_out_kernel<<<(ME * (FDE / 4)) / 256, 256, 0, stream>>>(
        edges, noise_edges, receivers, senders, counters, send_ids,
        out_edges, out_nrec, out_nsend, out_naedges);
}